// BitGAT_1932735283956
// MI455X (gfx1250) — compile-verified
//
#include <hip/hip_runtime.h>
#include <math.h>

#define NN   100000
#define EE   32768
#define DEGN 8
#define HH   256
#define TT   64
#define INCN (EE * DEGN)

typedef float v2f __attribute__((ext_vector_type(2)));
typedef float v8f __attribute__((ext_vector_type(8)));

__device__ __forceinline__ float waveReduceSum(float v) {
#pragma unroll
  for (int off = 16; off > 0; off >>= 1) v += __shfl_xor(v, off, 32);
  return v;
}

// Order-preserving float -> signed-int key (for atomicMax-based segment max).
__device__ __forceinline__ int f32_key(float f) {
  int b = __float_as_int(f);
  return (b >= 0) ? b : (b ^ 0x7fffffff);
}
__device__ __forceinline__ float key_f32(int k) {
  return __int_as_float((k >= 0) ? k : (k ^ 0x7fffffff));
}

// ---------------------------------------------------------------------------
// C[row, :] = (idx ? A[idx[row]] : A[row]) @ W + bias   (row-major, H=256)
// One wave computes a 16x16 f32 tile via V_WMMA_F32_16X16X4_F32.
// blockDim = 256 (8 waves); wave w covers column tiles {w, w+8}.
// grid.x = Mrows/16 (exact; M is a multiple of 16 -> EXEC all ones).
// ---------------------------------------------------------------------------
__global__ __launch_bounds__(256) void gemm_bias_wmma(
    const float* __restrict__ A, const int* __restrict__ idx,
    const float* __restrict__ W, const float* __restrict__ bias,
    float* __restrict__ C) {
  const int lane = threadIdx.x & 31;
  const int wave = threadIdx.x >> 5;
  const int rit  = lane & 15;   // row-in-tile (and col-in-tile for B/C)
  const int hi   = lane >> 4;   // K half select for A/B fragments
  const int row  = blockIdx.x * 16 + rit;
  const int arow = idx ? idx[row] : row;
  const float* Arow = A + (size_t)arow * HH;

#pragma unroll 1
  for (int ct = wave; ct < HH / 16; ct += 8) {
    const int ncol = ct * 16 + rit;
    v8f acc = {};
#pragma unroll 4
    for (int k0 = 0; k0 < HH; k0 += 4) {
      const int ka = k0 + hi * 2;
      v2f a;
      a.x = Arow[ka];
      a.y = Arow[ka + 1];
      v2f b;
      b.x = W[(size_t)ka * HH + ncol];
      b.y = W[(size_t)(ka + 1) * HH + ncol];
      acc = __builtin_amdgcn_wmma_f32_16x16x4_f32(
          false, a, false, b, (short)0, acc, false, false);
    }
    const float bv = bias[ncol];
    const int baseRow = blockIdx.x * 16 + hi * 8;
#pragma unroll
    for (int r = 0; r < 8; ++r)
      C[(size_t)(baseRow + r) * HH + ncol] = acc[r] + bv;
  }
}

// ---------------------------------------------------------------------------
// Edge aggregation with contiguous DEG=8 segments: one block per hyperedge.
// Wave j handles incidence e*8+j (logit via leaky_relu dot alpha), softmax of
// the 8 logits in LDS, then 256 threads write sign*(edge_emb[eidx[e]] + agg).
// aggr_out may alias pe (pe only read before the barrier, via LDS staging).
// ---------------------------------------------------------------------------
__global__ __launch_bounds__(256) void edge_attn_aggr(
    const float* __restrict__ edge_emb, const int* __restrict__ eidx,
    const float* __restrict__ pe, const float* __restrict__ proj_node,
    const float* __restrict__ node_emb, const float* __restrict__ alpha_w,
    const int* __restrict__ nidx_arr, const float sign,
    float* __restrict__ aggr_out) {
  __shared__ float s_pe[HH];
  __shared__ float s_logit[DEGN];
  __shared__ int   s_nd[DEGN];
  const int e = blockIdx.x;
  const int lane = threadIdx.x & 31;
  const int wave = threadIdx.x >> 5;
  s_pe[threadIdx.x] = pe[(size_t)e * HH + threadIdx.x];
  const int nd = nidx_arr[e * DEGN + wave];
  __syncthreads();

  const float* prow = proj_node + (size_t)nd * HH;
  float part = 0.f;
#pragma unroll
  for (int h = lane; h < HH; h += 32) {
    float v = s_pe[h] + prow[h];
    v = (v > 0.f) ? v : 0.01f * v;               // leaky_relu(0.01)
    part += v * alpha_w[h];
  }
  part = waveReduceSum(part);
  if (lane == 0) { s_logit[wave] = part; s_nd[wave] = nd; }
  __syncthreads();

  float m = s_logit[0];
#pragma unroll
  for (int j = 1; j < DEGN; ++j) m = fmaxf(m, s_logit[j]);
  float wgt[DEGN];
  float s = 0.f;
#pragma unroll
  for (int j = 0; j < DEGN; ++j) { wgt[j] = expf(s_logit[j] - m); s += wgt[j]; }
  const float inv = 1.f / s;

  const int h = threadIdx.x;
  float agg = 0.f;
#pragma unroll
  for (int j = 0; j < DEGN; ++j)
    agg += wgt[j] * node_emb[(size_t)s_nd[j] * HH + h];
  const float base = edge_emb[(size_t)eidx[e] * HH + h];
  aggr_out[(size_t)e * HH + h] = sign * (base + agg * inv);
}

// ---------------------------------------------------------------------------
// edge_emb_new scatter:  dst[dstidx[e], :] += src[e, :]
// ---------------------------------------------------------------------------
__global__ __launch_bounds__(256) void scatter_add_rows(
    const float* __restrict__ src, const int* __restrict__ dstidx,
    float* __restrict__ dst) {
  const int e = blockIdx.x;
  const int h = threadIdx.x;
  atomicAdd(&dst[(size_t)dstidx[e] * HH + h], src[(size_t)e * HH + h]);
}

// node time features: ntf[n,t] = cos(node_time[n]*w[t]+b[t])
__global__ __launch_bounds__(256) void node_tf_kernel(
    const float* __restrict__ node_time, const float* __restrict__ tw,
    const float* __restrict__ tb, float* __restrict__ ntf) {
  const int i = blockIdx.x * 256 + threadIdx.x;
  if (i >= NN * TT) return;
  const int n = i >> 6, t = i & (TT - 1);
  ntf[i] = cosf(node_time[n] * tw[t] + tb[t]);
}

// ---------------------------------------------------------------------------
// edge_propagate pass 1: logits per incidence + per-node running max (atomic
// on order-preserving int keys).  One block per edge, wave per incidence.
// ---------------------------------------------------------------------------
__global__ __launch_bounds__(256) void prop_logits(
    const float* __restrict__ edge_new, const int* __restrict__ eidx,
    const float* __restrict__ etime, const float* __restrict__ time_w,
    const float* __restrict__ time_b, const float* __restrict__ node_emb,
    const float* __restrict__ node_tf, const int* __restrict__ nidx_arr,
    float* __restrict__ logits, int* __restrict__ m_enc) {
  __shared__ float s_e[HH];
  __shared__ float s_tf[TT];
  const int e = blockIdx.x;
  const int lane = threadIdx.x & 31;
  const int wave = threadIdx.x >> 5;
  s_e[threadIdx.x] = edge_new[(size_t)eidx[e] * HH + threadIdx.x];
  if (threadIdx.x < TT) {
    const float t = etime[e];
    s_tf[threadIdx.x] = cosf(t * time_w[threadIdx.x] + time_b[threadIdx.x]);
  }
  const int i = e * DEGN + wave;
  const int nd = nidx_arr[i];
  __syncthreads();

  const float* nrow = node_emb + (size_t)nd * HH;
  const float* trow = node_tf + (size_t)nd * TT;
  float part = 0.f;
#pragma unroll
  for (int h = lane; h < HH; h += 32) part += s_e[h] * nrow[h];
#pragma unroll
  for (int t = lane; t < TT; t += 32) part += s_tf[t] * trow[t];
  part = waveReduceSum(part);
  if (lane == 0) {
    const float lg = part * 0.05590169943749474f;  // 1/sqrt(H+T)
    logits[i] = lg;
    atomicMax(&m_enc[nd], f32_key(lg));
  }
}

// pass 2: exp(logit - max[node]) and per-node sum
__global__ __launch_bounds__(256) void prop_exp(
    const float* __restrict__ logits, const int* __restrict__ nidx_arr,
    const int* __restrict__ m_enc, float* __restrict__ expv,
    float* __restrict__ ssum) {
  const int i = blockIdx.x * 256 + threadIdx.x;
  const int nd = nidx_arr[i];
  const float ev = expf(logits[i] - key_f32(m_enc[nd]));
  expv[i] = ev;
  atomicAdd(&ssum[nd], ev);
}

// pass 3: out[node, :] += sign * a_i * edge_new[eidx[e], :]
__global__ __launch_bounds__(256) void prop_scatter(
    const float* __restrict__ edge_new, const int* __restrict__ eidx,
    const int* __restrict__ nidx_arr, const float* __restrict__ expv,
    const float* __restrict__ ssum, const float sign,
    float* __restrict__ outp) {
  const int i = blockIdx.x;       // incidence
  const int e = i >> 3;
  const int nd = nidx_arr[i];
  const float a = sign * expv[i] / ssum[nd];
  const int h = threadIdx.x;
  atomicAdd(&outp[(size_t)nd * HH + h], a * edge_new[(size_t)eidx[e] * HH + h]);
}

// utilities -----------------------------------------------------------------
__global__ __launch_bounds__(256) void fill_i32(int* p, int v, int n) {
  const int i = blockIdx.x * 256 + threadIdx.x;
  if (i < n) p[i] = v;
}
__global__ __launch_bounds__(256) void fill_f32(float* p, float v, int n) {
  const int i = blockIdx.x * 256 + threadIdx.x;
  if (i < n) p[i] = v;
}
__global__ __launch_bounds__(256) void copy_f32(float* __restrict__ dst,
                                                const float* __restrict__ src,
                                                size_t n) {
  for (size_t i = (size_t)blockIdx.x * 256 + threadIdx.x; i < n;
       i += (size_t)gridDim.x * 256)
    dst[i] = src[i];
}

extern "C" void kernel_launch(void* const* d_in, const int* in_sizes, int n_in,
                              void* d_out, int out_size, void* d_ws,
                              size_t ws_size, hipStream_t stream) {
  (void)in_sizes; (void)n_in; (void)out_size; (void)ws_size;
  const float* node_emb      = (const float*)d_in[0];
  const float* edge_emb      = (const float*)d_in[1];
  const float* W_v           = (const float*)d_in[2];
  const float* b_v           = (const float*)d_in[3];
  const float* W_e           = (const float*)d_in[4];
  const float* b_e           = (const float*)d_in[5];
  const float* alpha_w       = (const float*)d_in[6];
  const float* time_w        = (const float*)d_in[7];
  const float* time_b        = (const float*)d_in[8];
  const float* edge_in_time  = (const float*)d_in[9];
  const float* edge_out_time = (const float*)d_in[10];
  const float* node_time     = (const float*)d_in[11];
  const int* in_edge_idx     = (const int*)d_in[12];
  const int* out_edge_idx    = (const int*)d_in[13];
  const int* in_node_idx     = (const int*)d_in[14];
  const int* out_node_idx    = (const int*)d_in[15];
  // d_in[16..17] (seg arrays) are implicit: contiguous runs of DEG.

  char* wptr = (char*)d_ws;
  auto alloc = [&](size_t bytes) -> char* {
    char* p = wptr;
    wptr += (bytes + 255) & ~(size_t)255;
    return p;
  };
  float* proj_node = (float*)alloc(sizeof(float) * (size_t)NN * HH);
  float* pe_in     = (float*)alloc(sizeof(float) * (size_t)EE * HH);  // -> in_aggr
  float* pe_out    = (float*)alloc(sizeof(float) * (size_t)EE * HH);  // -> out_aggr
  float* edge_new  = (float*)alloc(sizeof(float) * (size_t)EE * HH);
  float* ntf       = (float*)alloc(sizeof(float) * (size_t)NN * TT);
  float* logits    = (float*)alloc(sizeof(float) * INCN);
  float* expv      = (float*)alloc(sizeof(float) * INCN);
  int*   m_enc     = (int*)alloc(sizeof(int) * NN);
  float* ssum      = (float*)alloc(sizeof(float) * NN);
  float* out       = (float*)d_out;

  // --- GEMMs (f32 WMMA) ---
  gemm_bias_wmma<<<NN / 16, 256, 0, stream>>>(node_emb, nullptr, W_v, b_v,
                                              proj_node);
  gemm_bias_wmma<<<EE / 16, 256, 0, stream>>>(edge_emb, in_edge_idx, W_e, b_e,
                                              pe_in);
  gemm_bias_wmma<<<EE / 16, 256, 0, stream>>>(edge_emb, out_edge_idx, W_e, b_e,
                                              pe_out);

  // --- edge aggregation (segment softmax over contiguous DEG=8) ---
  edge_attn_aggr<<<EE, 256, 0, stream>>>(edge_emb, in_edge_idx, pe_in,
                                         proj_node, node_emb, alpha_w,
                                         in_node_idx, 1.f, pe_in);
  edge_attn_aggr<<<EE, 256, 0, stream>>>(edge_emb, out_edge_idx, pe_out,
                                         proj_node, node_emb, alpha_w,
                                         out_node_idx, -1.f, pe_out);

  // --- edge_emb_new = edge_emb + scatter(in_aggr) + scatter(out_aggr) ---
  copy_f32<<<2048, 256, 0, stream>>>(edge_new, edge_emb, (size_t)EE * HH);
  scatter_add_rows<<<EE, 256, 0, stream>>>(pe_in, in_edge_idx, edge_new);
  scatter_add_rows<<<EE, 256, 0, stream>>>(pe_out, out_edge_idx, edge_new);

  // --- propagation setup ---
  node_tf_kernel<<<(NN * TT + 255) / 256, 256, 0, stream>>>(node_time, time_w,
                                                            time_b, ntf);
  copy_f32<<<4096, 256, 0, stream>>>(out, node_emb, (size_t)NN * HH);

  for (int path = 0; path < 2; ++path) {
    const int* eidx = (path == 0) ? in_edge_idx : out_edge_idx;
    const int* nidx = (path == 0) ? in_node_idx : out_node_idx;
    const float* et = (path == 0) ? edge_in_time : edge_out_time;
    const float sign = (path == 0) ? 1.f : -1.f;
    fill_i32<<<(NN + 255) / 256, 256, 0, stream>>>(m_enc, (int)0x80000000u, NN);
    fill_f32<<<(NN + 255) / 256, 256, 0, stream>>>(ssum, 0.f, NN);
    prop_logits<<<EE, 256, 0, stream>>>(edge_new, eidx, et, time_w, time_b,
                                        node_emb, ntf, nidx, logits, m_enc);
    prop_exp<<<INCN / 256, 256, 0, stream>>>(logits, nidx, m_enc, expv, ssum);
    prop_scatter<<<INCN, 256, 0, stream>>>(edge_new, eidx, nidx, expv, ssum,
                                           sign, out);
  }
}